// Attention_28243704938858
// MI455X (gfx1250) — compile-verified
//
#include <hip/hip_runtime.h>
#include <math.h>

typedef float v2f __attribute__((ext_vector_type(2)));
typedef float v8f __attribute__((ext_vector_type(8)));

#define NCOL 16384   // B*T = 16*1024
#define TDIM 1024

// ---------------------------------------------------------------------------
// Kernel 1: GroupNorm (affine=False). One block per (batch, group).
// Group = 16 consecutive channels x 1024 = contiguous 16384 floats.
// Writes gn in [C][B*T] layout for the downstream GEMMs.
// ---------------------------------------------------------------------------
__global__ __launch_bounds__(256)
void gn_kernel(const float* __restrict__ x, float* __restrict__ gn) {
    __shared__ float ssum[256];
    __shared__ float ssq[256];
    const int b = blockIdx.x >> 4;
    const int g = blockIdx.x & 15;
    const float* xp = x + b * 262144 + g * 16384;

    float s = 0.f, q = 0.f;
    for (int i = threadIdx.x; i < 16384; i += 256) {
        float v = xp[i];
        s += v;
        q += v * v;
    }
    ssum[threadIdx.x] = s;
    ssq[threadIdx.x]  = q;
    __syncthreads();
    for (int off = 128; off > 0; off >>= 1) {
        if (threadIdx.x < off) {
            ssum[threadIdx.x] += ssum[threadIdx.x + off];
            ssq[threadIdx.x]  += ssq[threadIdx.x + off];
        }
        __syncthreads();
    }
    const float mean = ssum[0] * (1.f / 16384.f);
    const float var  = ssq[0] * (1.f / 16384.f) - mean * mean;
    const float rstd = rsqrtf(var + 1e-5f);

    for (int i = threadIdx.x; i < 16384; i += 256) {
        const int cl = i >> 10;          // channel within group
        const int t  = i & 1023;
        gn[(g * 16 + cl) * NCOL + b * TDIM + t] = (xp[i] - mean) * rstd;
    }
}

// ---------------------------------------------------------------------------
// Kernel 2/4: WMMA f32 GEMM:  Out[m][n] = sum_k W[m][k] * Bm[k][n] (+bias)
//   W: [M x K] row-major, Bm: [K x NCOL].
//   Each wave: 16(M) x 64(N) tile; block = 8 waves = 128 M rows.
//   mode 0: store to [M][NCOL]; rows m < scale_below scaled by scaleval
//           (folds the attention softmax scale into q,k at QKV time).
//   mode 1: scatter to [B,C,T] with residual add (output projection).
// ---------------------------------------------------------------------------
__global__ __launch_bounds__(256)
void gemm_kernel(const float* __restrict__ W, const float* __restrict__ Bm,
                 const float* __restrict__ bias, float* __restrict__ Out,
                 const float* __restrict__ resid,
                 int K, int scale_below, float scaleval, int mode) {
    const int tid  = threadIdx.x;
    const int wv   = tid >> 5;
    const int lane = tid & 31;
    const int col  = lane & 15;
    const int half = lane >> 4;
    const int kb   = half * 2;

    const int m0 = blockIdx.y * 128 + wv * 16;
    const int n0 = blockIdx.x * 64;

    v8f acc[4] = {v8f{}, v8f{}, v8f{}, v8f{}};

    const float* wrow = W + (m0 + col) * K;
    for (int k = 0; k < K; k += 4) {
        v2f a;
        a.x = wrow[k + kb];
        a.y = wrow[k + kb + 1];
#pragma unroll
        for (int j = 0; j < 4; ++j) {
            const float* bp = Bm + (k + kb) * NCOL + n0 + j * 16 + col;
            v2f bf;
            bf.x = bp[0];
            bf.y = bp[NCOL];
            acc[j] = __builtin_amdgcn_wmma_f32_16x16x4_f32(
                false, a, false, bf, (short)0, acc[j], false, false);
        }
    }

#pragma unroll
    for (int r = 0; r < 8; ++r) {
        const int m = m0 + r + 8 * half;
        const float mult = (m < scale_below) ? scaleval : 1.0f;
        const float bb = bias[m];
#pragma unroll
        for (int j = 0; j < 4; ++j) {
            const int n = n0 + j * 16 + col;
            float val = (acc[j][r] + bb) * mult;
            if (mode == 0) {
                Out[m * NCOL + n] = val;
            } else {
                const int flat = (n >> 10) * 262144 + m * TDIM + (n & 1023);
                Out[flat] = val + resid[flat];
            }
        }
    }
}

// ---------------------------------------------------------------------------
// Kernel 3: flash attention per (b, head). head_dim=32, T=1024.
//   qkv layout: [768][NCOL]; rows 0-255=q (pre-scaled), 256-511=k (pre-scaled),
//   512-767=v. Block = 8 waves; each wave owns 16 queries; all waves share the
//   same K/V chunk staged in LDS. Online softmax; output a in [C][NCOL].
// ---------------------------------------------------------------------------
__global__ __launch_bounds__(256)
void attn_kernel(const float* __restrict__ qkv, float* __restrict__ amem) {
    __shared__ float lds_k[32 * 65];       // padded: stride 65 avoids bank conflicts
    __shared__ float lds_v[32 * 65];
    __shared__ float lds_p[8 * 16 * 17];   // per-wave P tile, padded stride 17

    const int tid  = threadIdx.x;
    const int wv   = tid >> 5;
    const int lane = tid & 31;
    const int col  = lane & 15;
    const int half = lane >> 4;
    const int kb   = half * 2;

    const int bh = blockIdx.y;
    const int b  = bh >> 3;
    const int h  = bh & 7;
    const int t0 = blockIdx.x * 128 + wv * 16;

    const float* qp = qkv + (h * 32) * NCOL + b * TDIM;
    const float* kp = qkv + (256 + h * 32) * NCOL + b * TDIM;
    const float* vp = qkv + (512 + h * 32) * NCOL + b * TDIM;

    // Preload Q A-frags: A[M=t][K=ch], 8 k-steps of 4 over ch=32
    v2f qa[8];
#pragma unroll
    for (int kk = 0; kk < 8; ++kk) {
        const int ch = kk * 4 + kb;
        qa[kk].x = qp[ch * NCOL + t0 + col];
        qa[kk].y = qp[(ch + 1) * NCOL + t0 + col];
    }

    float mrun[8], lrun[8];
    v8f acc0 = {}, acc1 = {};
#pragma unroll
    for (int r = 0; r < 8; ++r) { mrun[r] = -INFINITY; lrun[r] = 0.f; }

    float* myP = &lds_p[wv * 16 * 17];

    for (int sc = 0; sc < TDIM; sc += 64) {
        __syncthreads();
        // cooperative stage of K,V chunk [32ch x 64s] into LDS
#pragma unroll
        for (int i = 0; i < 8; ++i) {
            const int idx = tid + i * 256;   // 0..2047
            const int ch = idx >> 6, sl = idx & 63;
            lds_k[ch * 65 + sl] = kp[ch * NCOL + sc + sl];
            lds_v[ch * 65 + sl] = vp[ch * NCOL + sc + sl];
        }
        __syncthreads();

#pragma unroll
        for (int st = 0; st < 4; ++st) {
            const int sl0 = st * 16;
            // ---- scores S = Q * K^T : 8 wmma over ch ----
            v8f s = {};
#pragma unroll
            for (int kk = 0; kk < 8; ++kk) {
                const int ch = kk * 4 + kb;
                v2f bf;
                bf.x = lds_k[ch * 65 + sl0 + col];
                bf.y = lds_k[(ch + 1) * 65 + sl0 + col];
                s = __builtin_amdgcn_wmma_f32_16x16x4_f32(
                    false, qa[kk], false, bf, (short)0, s, false, false);
            }
            // ---- online softmax (rows live in 16-lane halves) ----
            float p[8];
#pragma unroll
            for (int r = 0; r < 8; ++r) {
                float rv = s[r];
                float rmax = rv;
#pragma unroll
                for (int off = 1; off < 16; off <<= 1)
                    rmax = fmaxf(rmax, __shfl_xor(rmax, off, 32));
                const float mnew  = fmaxf(mrun[r], rmax);
                const float alpha = __expf(mrun[r] - mnew);
                mrun[r] = mnew;
                const float pv = __expf(rv - mnew);
                p[r] = pv;
                float rs = pv;
#pragma unroll
                for (int off = 1; off < 16; off <<= 1)
                    rs += __shfl_xor(rs, off, 32);
                lrun[r] = lrun[r] * alpha + rs;
                acc0[r] *= alpha;
                acc1[r] *= alpha;
            }
            // ---- P through LDS: D-layout -> A-layout (wave-local) ----
#pragma unroll
            for (int r = 0; r < 8; ++r)
                myP[(r + 8 * half) * 17 + col] = p[r];
            asm volatile("" ::: "memory");   // keep LDS store->load order
            // ---- AV: acc += P * V^T, K=16 s-steps, two c tiles ----
#pragma unroll
            for (int kk2 = 0; kk2 < 4; ++kk2) {
                v2f pa;
                pa.x = myP[col * 17 + kk2 * 4 + kb];
                pa.y = myP[col * 17 + kk2 * 4 + kb + 1];
                const int sl = sl0 + kk2 * 4 + kb;
                v2f vb0, vb1;
                vb0.x = lds_v[col * 65 + sl];
                vb0.y = lds_v[col * 65 + sl + 1];
                vb1.x = lds_v[(16 + col) * 65 + sl];
                vb1.y = lds_v[(16 + col) * 65 + sl + 1];
                acc0 = __builtin_amdgcn_wmma_f32_16x16x4_f32(
                    false, pa, false, vb0, (short)0, acc0, false, false);
                acc1 = __builtin_amdgcn_wmma_f32_16x16x4_f32(
                    false, pa, false, vb1, (short)0, acc1, false, false);
            }
            asm volatile("" ::: "memory");   // AV reads done before next P store
        }
    }

    // epilogue: normalize and store a in [C][NCOL]
#pragma unroll
    for (int r = 0; r < 8; ++r) {
        const float inv = 1.0f / lrun[r];
        const int t = t0 + r + 8 * half;
        const int n = b * TDIM + t;
        amem[(h * 32 + col) * NCOL + n]      = acc0[r] * inv;
        amem[(h * 32 + 16 + col) * NCOL + n] = acc1[r] * inv;
    }
}

// ---------------------------------------------------------------------------
extern "C" void kernel_launch(void* const* d_in, const int* in_sizes, int n_in,
                              void* d_out, int out_size, void* d_ws, size_t ws_size,
                              hipStream_t stream) {
    const float* x     = (const float*)d_in[0];   // [16,256,32,32]
    const float* qkv_w = (const float*)d_in[1];   // [768,256]
    const float* qkv_b = (const float*)d_in[2];   // [768]
    const float* out_w = (const float*)d_in[3];   // [256,256]
    const float* out_b = (const float*)d_in[4];   // [256]
    float* out = (float*)d_out;

    float* gn  = (float*)d_ws;                    // [256][16384]
    float* qkv = gn + 256 * NCOL;                 // [768][16384]
    float* a   = qkv + 768 * NCOL;                // [256][16384]

    const float scale = 0.42044820762685725f;     // 32^(-1/4)

    // 1) GroupNorm -> gn [C][B*T]
    gn_kernel<<<256, 256, 0, stream>>>(x, gn);
    // 2) QKV GEMM (q,k rows pre-scaled): [768x256] x [256x16384]
    gemm_kernel<<<dim3(256, 6), 256, 0, stream>>>(
        qkv_w, gn, qkv_b, qkv, nullptr, 256, 512, scale, 0);
    // 3) Flash attention -> a [C][B*T]
    attn_kernel<<<dim3(8, 128), 256, 0, stream>>>(qkv, a);
    // 4) Output projection + bias + residual -> out [B,C,H,W]
    gemm_kernel<<<dim3(256, 2), 256, 0, stream>>>(
        out_w, a, out_b, out, x, 256, 0, 1.0f, 1);
}